// AEVComputer_46789373723025
// MI455X (gfx1250) — compile-verified
//
#include <hip/hip_runtime.h>

// ---------------------------------------------------------------------------
// ANI AEV on gfx1250 (MI455X).
//  grid  = M*A blocks (one per center atom), block = 128 threads = 4 wave32.
//  Phase 1: stage coords/species, build j<k pair table.
//  Phase 2: per-neighbor geometry (dist, cutoffs) in LDS.
//  Phase 3: per-triple commons (cos/sin theta, rmean, 2*fcp, pair bin)
//           computed ONCE per block into LDS (no 16-way lane redundancy).
//  Phase 4: one-hot contractions via V_WMMA_F32_16X16X4_F32 — the
//           reference's einsum('...f,...s->...sf', terms, onehot) IS this GEMM.
//  Phase 5: cross-wave reduction with ds_add_f32, store 384 feats/atom.
// ---------------------------------------------------------------------------

typedef __attribute__((ext_vector_type(2))) float v2f;
typedef __attribute__((ext_vector_type(8))) float v8f;

#define A_ATOMS   40
#define NPAIRT    780            // 40*39/2 j<k pairs
#define NGRP_ANG  195            // 780/4  (K=4 per WMMA)
#define NGRP_RAD  10             // 40/4
#define RCR_      5.2f
#define RCA_      3.5f

__global__ __launch_bounds__(128)
void aev_wmma_kernel(const int* __restrict__ elem,
                     const float* __restrict__ coords,
                     float* __restrict__ out)
{
    __shared__ float   cx[A_ATOMS], cy[A_ATOMS], cz_[A_ATOMS];
    __shared__ float4  pj[A_ATOMS];        // (dx,dy,dz,d) relative to center i
    __shared__ float   fcRm[A_ATOMS];      // fc_r * radial mask
    __shared__ float   fcAm[A_ATOMS];      // fc_a * angular mask
    __shared__ int     spec[A_ATOMS];
    __shared__ unsigned short jk[NPAIRT];  // packed j | (k<<8), j<k
    __shared__ float4  tri[NPAIRT];        // (cos_t, sin_t, rmean, 2*fcp)
    __shared__ unsigned char tpair[NPAIRT];// species-pair bin 0..9
    __shared__ float   binsRad[64];        // [species 4][16 shells]
    __shared__ float   binsAng[320];       // [pair 10][32 feats]

    const int tid = threadIdx.x;
    const int blk = blockIdx.x;
    const int m   = blk / A_ATOMS;
    const int i   = blk % A_ATOMS;

    // ---- phase 1: stage coords/species, zero bins, build j<k pair table ----
    if (tid < A_ATOMS) {
        const float* c = coords + (size_t)(m * A_ATOMS + tid) * 3;
        cx[tid] = c[0]; cy[tid] = c[1]; cz_[tid] = c[2];
        spec[tid] = elem[m * A_ATOMS + tid];
    }
    for (int idx = tid; idx < 64;  idx += 128) binsRad[idx] = 0.0f;
    for (int idx = tid; idx < 320; idx += 128) binsAng[idx] = 0.0f;
    for (int idx = tid; idx < NPAIRT; idx += 128) {
        // unrank linear index -> (j<k) pair; off(j) = j*39 - j*(j-1)/2
        int j = (int)(39.5f - sqrtf(39.5f * 39.5f - 2.0f * (float)idx));
        if (j < 0) j = 0; if (j > 38) j = 38;
        while (j > 0  && (j * 39 - (j * (j - 1)) / 2) > idx) --j;
        while (j < 38 && (((j + 1) * 39 - ((j + 1) * j) / 2) <= idx)) ++j;
        int k = idx - (j * 39 - (j * (j - 1)) / 2) + j + 1;
        jk[idx] = (unsigned short)(j | (k << 8));
    }
    __syncthreads();

    // ---- phase 2: per-neighbor geometry relative to center i ----
    if (tid < A_ATOMS) {
        float dx = cx[tid] - cx[i];
        float dy = cy[tid] - cy[i];
        float dz = cz_[tid] - cz_[i];
        float d  = sqrtf(dx * dx + dy * dy + dz * dz + 1e-12f);
        pj[tid]  = make_float4(dx, dy, dz, d);
        float mR = (tid != i && d < RCR_) ? 1.0f : 0.0f;
        float mA = (tid != i && d < RCA_) ? 1.0f : 0.0f;
        fcRm[tid] = (0.5f * __cosf(3.14159265358979f / RCR_ * d) + 0.5f) * mR;
        fcAm[tid] = (0.5f * __cosf(3.14159265358979f / RCA_ * d) + 0.5f) * mA;
    }
    __syncthreads();

    // ---- phase 3: per-triple commons computed once per block ----
    for (int t = tid; t < NPAIRT; t += 128) {
        const int e = jk[t];
        const int j = e & 255, k = e >> 8;
        const float4 J = pj[j], K4 = pj[k];
        const float dj = J.w, dk = K4.w;
        const float dot = J.x * K4.x + J.y * K4.y + J.z * K4.z;
        const float ca  = 0.95f * dot / (dj * dk);          // cos(theta)
        float s2 = 1.0f - ca * ca;  s2 = (s2 < 0.f) ? 0.f : s2;
        const float sn  = sqrtf(s2);                        // sin(theta) >= 0
        tri[t] = make_float4(ca, sn, 0.5f * (dj + dk),
                             2.0f * fcAm[j] * fcAm[k]);     // masks folded
        const int sj = spec[j], sk = spec[k];
        const int mn = sj < sk ? sj : sk;
        const int mx = sj < sk ? sk : sj;
        tpair[t] = (unsigned char)(mn * (9 - mn) / 2 + (mx - mn));
    }
    __syncthreads();

    // ---- per-lane invariants (WMMA operand layouts, wave32) ----
    const int lane = tid & 31;
    const int w    = tid >> 5;            // wave id 0..3
    const int f    = lane & 15;           // N column / M row index
    const int hi   = lane >> 4;           // half-wave selects K slots {0,1} vs {2,3}
    const int ks0  = hi ? 2 : 0;          // K slot held in vector element .x
    const int ks1  = ks0 + 1;             // K slot held in vector element .y
    const float shfr = 0.9f + 0.26875f * (float)f;          // radial shell
    const int   iz = f & 7, ia = f >> 3;
    const float za = 0.19634954084936207f * (float)(2 * iz + 1);  // pi/16*(2z+1)
    const float czc = __cosf(za), szc = __sinf(za);
    const float sa_lo = 0.9f + 0.65f * (float)ia;           // shf_a[ia]
    const float sa_hi = 2.2f + 0.65f * (float)ia;           // shf_a[ia+2]

    // ===== radial: bins[s][f] = sum_j onehot[s][j] * term[j][f]  (K=4/WMMA) ==
    v8f cRad = {0.f,0.f,0.f,0.f,0.f,0.f,0.f,0.f};
    for (int g = w; g < NGRP_RAD; g += 4) {
        const int j0 = 4 * g + ks0, j1 = 4 * g + ks1;
        const float d0 = pj[j0].w, d1 = pj[j1].w;
        const float e0 = d0 - shfr, e1 = d1 - shfr;
        v2f a, b;
        b.x = 0.25f * __expf(-16.0f * e0 * e0) * fcRm[j0];
        b.y = 0.25f * __expf(-16.0f * e1 * e1) * fcRm[j1];
        a.x = (spec[j0] == f) ? 1.0f : 0.0f;    // onehot row = species
        a.y = (spec[j1] == f) ? 1.0f : 0.0f;
        cRad = __builtin_amdgcn_wmma_f32_16x16x4_f32(
            false, a, false, b, (short)0, cRad, false, false);
    }

    // ===== angular: bins[p][f] = sum_t onehot[p][t] * term[t][f] ============
    v8f c0 = {0.f,0.f,0.f,0.f,0.f,0.f,0.f,0.f};   // feature columns 0..15
    v8f c1 = {0.f,0.f,0.f,0.f,0.f,0.f,0.f,0.f};   // feature columns 16..31
    for (int g = w; g < NGRP_ANG; g += 4) {
        const int t0 = 4 * g + ks0, t1 = 4 * g + ks1;
        float b0x, b0y, b1x, b1y;
        #pragma unroll
        for (int s = 0; s < 2; ++s) {
            const float4 T = tri[s ? t1 : t0];
            // cos(theta - shf_z[iz]) via angle-difference identity (no acos)
            const float cd = T.x * czc + T.y * szc;
            float pz = 0.5f * (1.0f + cd);                  // ^32 = 5 squarings
            pz *= pz; pz *= pz; pz *= pz; pz *= pz; pz *= pz;
            const float g2 = pz * T.w;                      // 2*fcp folded in T.w
            const float dl = T.z - sa_lo, dh = T.z - sa_hi;
            const float tlo = g2 * __expf(-8.0f * dl * dl);
            const float thi = g2 * __expf(-8.0f * dh * dh);
            if (s == 0) { b0x = tlo; b1x = thi; }
            else        { b0y = tlo; b1y = thi; }
        }
        v2f a, b0, b1;
        a.x = ((int)tpair[t0] == f) ? 1.0f : 0.0f;   // onehot row = pair bin
        a.y = ((int)tpair[t1] == f) ? 1.0f : 0.0f;
        b0.x = b0x; b0.y = b0y;
        b1.x = b1x; b1.y = b1y;
        c0 = __builtin_amdgcn_wmma_f32_16x16x4_f32(
            false, a, false, b0, (short)0, c0, false, false);
        c1 = __builtin_amdgcn_wmma_f32_16x16x4_f32(
            false, a, false, b1, (short)0, c1, false, false);
    }

    // ---- reduce the 4 waves' D tiles into LDS bins (C/D: VGPR v = rows v, v+8)
    #pragma unroll
    for (int v = 0; v < 8; ++v) {
        const int row = hi ? (v + 8) : v;
        if (!hi && v < 4) atomicAdd(&binsRad[v * 16 + f], cRad[v]);
        if (row < 10) {
            atomicAdd(&binsAng[row * 32 + f],      c0[v]);
            atomicAdd(&binsAng[row * 32 + 16 + f], c1[v]);
        }
    }
    __syncthreads();

    // ---- write 64 radial + 320 angular features for this (m,i) ----
    float* o = out + (size_t)blk * 384;
    for (int idx = tid; idx < 384; idx += 128)
        o[idx] = (idx < 64) ? binsRad[idx] : binsAng[idx - 64];
}

extern "C" void kernel_launch(void* const* d_in, const int* in_sizes, int n_in,
                              void* d_out, int out_size, void* d_ws, size_t ws_size,
                              hipStream_t stream)
{
    const int*   elem   = (const int*)d_in[0];    // (M, 40) int32
    const float* coords = (const float*)d_in[1];  // (M, 40, 3) f32
    float*       out    = (float*)d_out;          // (M, 40, 384) f32
    const int MA = in_sizes[0];                   // M * 40 center atoms
    aev_wmma_kernel<<<MA, 128, 0, stream>>>(elem, coords, out);
    (void)n_in; (void)out_size; (void)d_ws; (void)ws_size;
}